// ClusterNet_26517128085842
// MI455X (gfx1250) — compile-verified
//
#include <hip/hip_runtime.h>

// ============================================================================
// ClusterNet forward for MI455X (gfx1250, wave32, WMMA f32 16x16x4)
// - All matmul-shaped work uses V_WMMA_F32_16X16X4_F32 (fp32 net, HBM-bound).
// - GEMM stages weight panels in LDS (ds_load B-operand, b64 A-operand loads)
//   and can fuse the previous layer's BN affine into its A-loads, removing
//   full-tensor bn_apply passes between chained layers.
// - Dense adjacency never materialized: mincut terms use the sparse kNN list.
// ============================================================================

typedef float v2f __attribute__((ext_vector_type(2)));
typedef float v8f __attribute__((ext_vector_type(8)));

#define WMMA_F32_4(a, b, c) \
  __builtin_amdgcn_wmma_f32_16x16x4_f32(false, (a), false, (b), (short)0, (c), false, false)

// ---------------------------------------------------------------------------
// elementwise / utility kernels
// ---------------------------------------------------------------------------
__global__ void cn_zero(float* __restrict__ p, size_t n) {
  size_t t = (size_t)blockIdx.x * blockDim.x + threadIdx.x;
  if (t < n) p[t] = 0.f;
}

__global__ void cn_sqnorm(const float* __restrict__ x, float* __restrict__ sq,
                          int total, int C) {
  int t = blockIdx.x * blockDim.x + threadIdx.x;
  if (t >= total) return;
  const float* p = x + (size_t)t * C;
  float a = 0.f;
  for (int c = 0; c < C; ++c) a += p[c] * p[c];
  sq[t] = a;
}

// ---------------------------------------------------------------------------
// Stage-1 kNN: 16x16 distance tiles via WMMA f32 16x16x4 (K=3 padded to 4)
// D[i][j] = |xi|^2 + |xj|^2 - 2 xi.xj  (+1e10 on the diagonal)
// ---------------------------------------------------------------------------
__global__ void cn_dist_wmma(const float* __restrict__ x, const float* __restrict__ sq,
                             float* __restrict__ dist, int b, int N) {
  const int lane = threadIdx.x, half = lane >> 4, l16 = lane & 15;
  const int i0 = blockIdx.y * 16, j0 = blockIdx.x * 16;
  const float* xb = x + (size_t)b * N * 3;
  const float* sqb = sq + (size_t)b * N;
  const int ka = 2 * half;  // K index 0 or 2 (K padded 3->4)
  v2f a, bb;
  a.x = xb[(size_t)(i0 + l16) * 3 + ka];
  a.y = (ka + 1 < 3) ? xb[(size_t)(i0 + l16) * 3 + ka + 1] : 0.f;
  bb.x = xb[(size_t)(j0 + l16) * 3 + ka];
  bb.y = (ka + 1 < 3) ? xb[(size_t)(j0 + l16) * 3 + ka + 1] : 0.f;
  v8f acc = {};
  acc = WMMA_F32_4(a, bb, acc);
  const int j = j0 + l16;
  const float sqj = sqb[j];
#pragma unroll
  for (int r = 0; r < 8; ++r) {
    int i = i0 + r + 8 * half;
    float d = sqb[i] + sqj - 2.f * acc[r];
    if (i == j) d += 1e10f;
    dist[(size_t)i * N + j] = d;
  }
}

// One wave per row: each lane keeps a sorted local top-16, then a 32-way merge.
__global__ void cn_topk16(const float* __restrict__ dist, int* __restrict__ nbr,
                          int b, int N, int K) {
  const int row = blockIdx.x;
  const int lane = threadIdx.x;
  float v[16];
  int id[16];
#pragma unroll
  for (int t = 0; t < 16; ++t) { v[t] = 3.4e38f; id[t] = -1; }
  const float* drow = dist + (size_t)row * N;
  for (int j = lane; j < N; j += 32) {
    float d = drow[j];
    if (d < v[15]) {
      v[15] = d; id[15] = j;
#pragma unroll
      for (int t = 15; t > 0; --t) {
        if (v[t] < v[t - 1]) {
          float tv = v[t]; v[t] = v[t - 1]; v[t - 1] = tv;
          int ti = id[t]; id[t] = id[t - 1]; id[t - 1] = ti;
        }
      }
    }
  }
  __shared__ float sv[512];
  __shared__ int sid[512];
#pragma unroll
  for (int t = 0; t < 16; ++t) { sv[lane * 16 + t] = v[t]; sid[lane * 16 + t] = id[t]; }
  __syncthreads();
  if (lane == 0) {
    int ptr[32];
    for (int l = 0; l < 32; ++l) ptr[l] = 0;
    int* out = nbr + ((size_t)b * N + row) * K;
    for (int t = 0; t < K; ++t) {
      float best = 3.5e38f; int bl = 0;
      for (int l = 0; l < 32; ++l) {
        float cv = sv[l * 16 + ptr[l]];
        if (cv < best) { best = cv; bl = l; }
      }
      out[t] = sid[bl * 16 + ptr[bl]];
      ptr[bl]++;
    }
  }
}

// Small kNN (n<=32) for stage 2: one block per batch.
__global__ void cn_knn_small(const float* __restrict__ x, int* __restrict__ nbr,
                             int n, int C, int K) {
  const int b = blockIdx.x;
  __shared__ float d[32 * 32];
  const float* xb = x + (size_t)b * n * C;
  for (int p = threadIdx.x; p < n * n; p += blockDim.x) {
    int i = p / n, j = p % n;
    float acc = 0.f;
    for (int c = 0; c < C; ++c) { float t = xb[i * C + c] - xb[j * C + c]; acc += t * t; }
    d[p] = (i == j) ? 1e10f : acc;
  }
  __syncthreads();
  if (threadIdx.x < (unsigned)n) {
    const int i = threadIdx.x;
    bool used[32];
    for (int j = 0; j < n; ++j) used[j] = false;
    int* out = nbr + ((size_t)b * n + i) * K;
    for (int t = 0; t < K; ++t) {
      float best = 3.5e38f; int bj = 0;
      for (int j = 0; j < n; ++j)
        if (!used[j] && d[i * n + j] < best) { best = d[i * n + j]; bj = j; }
      used[bj] = true;
      out[t] = bj;
    }
  }
}

// Stage 3: k = n-1 -> neighbors are simply all other nodes.
__global__ void cn_nbr_all(int* __restrict__ nbr, int Bc, int n, int K) {
  int t = blockIdx.x * blockDim.x + threadIdx.x;
  if (t >= Bc * n * K) return;
  int k = t % K, i = (t / K) % n;
  nbr[t] = (k < i) ? k : k + 1;
}

// ---------------------------------------------------------------------------
// EdgeConv plumbing
// ---------------------------------------------------------------------------
__global__ void cn_edge_feat(const float* __restrict__ x, const int* __restrict__ nbr,
                             float* __restrict__ ef, int Bc, int n, int K, int C) {
  size_t t = (size_t)blockIdx.x * blockDim.x + threadIdx.x;
  size_t total = (size_t)Bc * n * K * 2 * C;
  if (t >= total) return;
  int c = (int)(t % (2 * C));
  size_t e = t / (2 * C);
  size_t bi = e / K;
  int i = (int)(bi % n);
  int b = (int)(bi / n);
  const float* xb = x + ((size_t)b * n) * C;
  int j = nbr[e];
  float val;
  if (c < C) val = xb[(size_t)i * C + c];
  else { int cc = c - C; val = xb[(size_t)j * C + cc] - xb[(size_t)i * C + cc]; }
  ef[t] = val;
}

__global__ void cn_edge_max(const float* __restrict__ Y, float* __restrict__ out,
                            int Bc, int n, int K, int C) {
  size_t t = (size_t)blockIdx.x * blockDim.x + threadIdx.x;
  size_t total = (size_t)Bc * n * C;
  if (t >= total) return;
  int c = (int)(t % C);
  size_t bi = t / C;
  const float* row = Y + bi * (size_t)K * C + c;
  float m = row[0];
  for (int kk = 1; kk < K; ++kk) m = fmaxf(m, row[(size_t)kk * C]);
  out[t] = m;
}

// ---------------------------------------------------------------------------
// Generic WMMA GEMM: Y[R,Fout] = act((inBN(X))[R,Fin] @ W[Fin,Fout] + b)
//   inScsh: optional fused input BN affine (scale[0..Fin), shift[Fin..2Fin)).
//   W panel (K-panels of <=512) staged in LDS; A loaded as aligned float2.
//   Optional ReLU + per-channel BN stats (sum, sumsq) via global atomics.
// One wave per block; MT row-tiles per block amortize staging + atomics.
// ---------------------------------------------------------------------------
template <bool RELU, bool STATS>
__global__ void cn_gemm(const float* __restrict__ X, const float* __restrict__ inScsh,
                        const float* __restrict__ W, const float* __restrict__ bias,
                        float* __restrict__ Y, float* __restrict__ stats,
                        int R, int Fin, int Fout, int MT) {
  extern __shared__ float lds[];
  const int lane = threadIdx.x;
  const int half = lane >> 4, l16 = lane & 15;
  const int n0 = blockIdx.x * 16;
  const int nc = n0 + l16;
  const int Fin4 = (Fin + 3) & ~3;
  const int KP = (Fin4 < 512) ? Fin4 : 512;
  float* ldsS = lds;              // [2*Fin4]: scale | shift (identity if !inScsh)
  float* ldsW = lds + 2 * Fin4;   // [KP*16]

  for (int u = lane; u < Fin4; u += 32) {
    float sc = 1.f, sh = 0.f;
    if (inScsh) { sc = (u < Fin) ? inScsh[u] : 0.f; sh = (u < Fin) ? inScsh[Fin + u] : 0.f; }
    else if (u >= Fin) sc = 0.f;
    ldsS[u] = sc;
    ldsS[Fin4 + u] = sh;
  }

  const int mbase = blockIdx.y * MT * 16;
  const bool evenFin = (Fin & 1) == 0;
  float ssum = 0.f, ssq = 0.f;
  bool staged = false;
  for (int t = 0; t < MT; ++t) {
    const int m0 = mbase + t * 16;
    if (m0 >= R) break;
    const int mrow_a = m0 + l16;
    const float* xrow = X + (size_t)mrow_a * Fin;
    v8f acc = {};
    for (int kp = 0; kp < Fin4; kp += KP) {
      int kend = kp + KP;
      if (kend > Fin4) kend = Fin4;
      if (!staged || Fin4 > KP) {
        __syncthreads();
        for (int u = lane; u < (kend - kp) * 16; u += 32) {
          int k = kp + (u >> 4), col = n0 + (u & 15);
          ldsW[u] = (k < Fin && col < Fout) ? W[(size_t)k * Fout + col] : 0.f;
        }
        __syncthreads();
        staged = true;
      }
      for (int k0 = kp; k0 < kend; k0 += 4) {
        const int ka = k0 + 2 * half;
        v2f a, b;
        b.x = ldsW[(ka - kp) * 16 + l16];
        b.y = ldsW[(ka + 1 - kp) * 16 + l16];
        float ax = 0.f, ay = 0.f;
        if (mrow_a < R) {
          if (evenFin && ka + 1 < Fin) {
            float2 t2 = *(const float2*)(xrow + ka);   // 8B-aligned: Fin even, ka even
            ax = t2.x; ay = t2.y;
          } else {
            if (ka < Fin) ax = xrow[ka];
            if (ka + 1 < Fin) ay = xrow[ka + 1];
          }
          ax = ax * ldsS[ka] + ldsS[Fin4 + ka];
          ay = ay * ldsS[ka + 1] + ldsS[Fin4 + ka + 1];
        }
        a.x = ax; a.y = ay;
        acc = WMMA_F32_4(a, b, acc);
      }
    }
    if (nc < Fout) {
      const float bv = bias[nc];
#pragma unroll
      for (int r = 0; r < 8; ++r) {
        const int mrow = m0 + r + 8 * half;
        if (mrow < R) {
          float v = acc[r] + bv;
          if (RELU) v = fmaxf(v, 0.f);
          Y[(size_t)mrow * Fout + nc] = v;
          if (STATS) { ssum += v; ssq += v * v; }
        }
      }
    }
  }
  if (STATS && nc < Fout) {
    atomicAdd(&stats[nc], ssum);
    atomicAdd(&stats[Fout + nc], ssq);
  }
}

__global__ void cn_bn_finalize(const float* __restrict__ stats, const float* __restrict__ g,
                               const float* __restrict__ beta, float* __restrict__ scsh,
                               int Fout, float invCount) {
  int c = blockIdx.x * blockDim.x + threadIdx.x;
  if (c >= Fout) return;
  float mean = stats[c] * invCount;
  float var = stats[Fout + c] * invCount - mean * mean;
  float sc = g[c] * rsqrtf(var + 1e-5f);
  scsh[c] = sc;
  scsh[Fout + c] = beta[c] - mean * sc;
}

__global__ void cn_bn_apply(float* __restrict__ Y, const float* __restrict__ scsh,
                            size_t total, int Fout) {
  size_t t = (size_t)blockIdx.x * blockDim.x + threadIdx.x;
  if (t >= total) return;
  int c = (int)(t % Fout);
  Y[t] = Y[t] * scsh[c] + scsh[Fout + c];
}

// ---------------------------------------------------------------------------
// Mincut pooling pieces (adjacency kept sparse via the kNN edge list)
// ---------------------------------------------------------------------------
__global__ void cn_softmax_rows(float* __restrict__ s, int rows, int C) {
  int r = blockIdx.x * blockDim.x + threadIdx.x;
  if (r >= rows) return;
  float* p = s + (size_t)r * C;
  float m = p[0];
  for (int c = 1; c < C; ++c) m = fmaxf(m, p[c]);
  float sum = 0.f;
  for (int c = 0; c < C; ++c) { float e = expf(p[c] - m); p[c] = e; sum += e; }
  float inv = 1.f / sum;
  for (int c = 0; c < C; ++c) p[c] *= inv;
}

// out[b,c,f] = sum_n s[b,n,c] * x[b,n,f]   (s^T x, WMMA over K=n)
__global__ void cn_pool_stx(const float* __restrict__ s, const float* __restrict__ x,
                            float* __restrict__ out, int n, int Cc, int F) {
  const int b = blockIdx.z;
  const int lane = threadIdx.x, half = lane >> 4, l16 = lane & 15;
  const int c0 = blockIdx.y * 16, f0 = blockIdx.x * 16;
  const int cc = c0 + l16, ff = f0 + l16;
  const float* sb = s + (size_t)b * n * Cc;
  const float* xb = x + (size_t)b * n * F;
  v8f acc = {};
  for (int k0 = 0; k0 < n; k0 += 4) {
    const int kk = k0 + 2 * half;
    v2f a, bb;
    a.x = (cc < Cc) ? sb[(size_t)kk * Cc + cc] : 0.f;        // A = s^T
    a.y = (cc < Cc) ? sb[(size_t)(kk + 1) * Cc + cc] : 0.f;
    bb.x = (ff < F) ? xb[(size_t)kk * F + ff] : 0.f;
    bb.y = (ff < F) ? xb[(size_t)(kk + 1) * F + ff] : 0.f;
    acc = WMMA_F32_4(a, bb, acc);
  }
  if (ff < F) {
#pragma unroll
    for (int r = 0; r < 8; ++r) {
      const int c = c0 + r + 8 * half;
      if (c < Cc) out[((size_t)b * Cc + c) * F + ff] = acc[r];
    }
  }
}

__global__ void cn_deg(const int* __restrict__ nbr, float* __restrict__ deg,
                       int Bc, int n, int K) {
  int t = blockIdx.x * blockDim.x + threadIdx.x;
  if (t >= Bc * n * K) return;
  int b = t / (n * K);
  atomicAdd(&deg[(size_t)b * n + nbr[t]], 1.0f);
}

// num[b] = trace(s^T A s) = sum over edges (j->i) of dot(s[j,:], s[i,:])
__global__ void cn_mincut_num(const float* __restrict__ s, const int* __restrict__ nbr,
                              float* __restrict__ num, int Bc, int n, int K, int Cc) {
  int t = blockIdx.x * blockDim.x + threadIdx.x;
  float acc = 0.f;
  int b = 0;
  if (t < Bc * n * K) {
    b = t / (n * K);
    int i = (t / K) % n;
    int j = nbr[t];
    const float* sb = s + (size_t)b * n * Cc;
    for (int c = 0; c < Cc; ++c) acc += sb[(size_t)j * Cc + c] * sb[(size_t)i * Cc + c];
  }
  for (int off = 16; off; off >>= 1) acc += __shfl_down(acc, off, 32);
  if ((threadIdx.x & 31) == 0 && t < Bc * n * K) atomicAdd(&num[b], acc);
}

// den[b] = sum_n deg[b,n] * sum_c s[b,n,c]^2
__global__ void cn_mincut_den(const float* __restrict__ s, const float* __restrict__ deg,
                              float* __restrict__ den, int Bc, int n, int Cc) {
  int t = blockIdx.x * blockDim.x + threadIdx.x;
  float acc = 0.f;
  int b = 0;
  if (t < Bc * n) {
    b = t / n;
    const float* sp = s + (size_t)t * Cc;
    for (int c = 0; c < Cc; ++c) acc += sp[c] * sp[c];
    acc *= deg[t];
  }
  for (int off = 16; off; off >>= 1) acc += __shfl_down(acc, off, 32);
  if ((threadIdx.x & 31) == 0 && t < Bc * n) atomicAdd(&den[b], acc);
}

// ss[b,c,d] = sum_n s[b,n,c] * s[b,n,d]
__global__ void cn_ss(const float* __restrict__ s, float* __restrict__ ss,
                      int Bc, int n, int Cc) {
  int t = blockIdx.x * blockDim.x + threadIdx.x;
  if (t >= Bc * Cc * Cc) return;
  int d = t % Cc, c = (t / Cc) % Cc, b = t / (Cc * Cc);
  const float* sb = s + (size_t)b * n * Cc;
  float acc = 0.f;
  for (int i = 0; i < n; ++i) acc += sb[(size_t)i * Cc + c] * sb[(size_t)i * Cc + d];
  ss[t] = acc;
}

__global__ void cn_mincut_loss(const float* __restrict__ num, const float* __restrict__ den,
                               const float* __restrict__ ss, float* __restrict__ accum,
                               int Bc, int Cc) {
  __shared__ float smc[32], sor[32];
  int b = threadIdx.x;
  float mc = 0.f, ortho = 0.f;
  if (b < Bc) {
    mc = -(num[b] / den[b]);
    const float* ssb = ss + (size_t)b * Cc * Cc;
    float fro = 0.f;
    for (int t = 0; t < Cc * Cc; ++t) fro += ssb[t] * ssb[t];
    float inv = rsqrtf(fro);
    float iden = rsqrtf((float)Cc);
    float acc = 0.f;
    for (int c = 0; c < Cc; ++c)
      for (int d = 0; d < Cc; ++d) {
        float vd = ssb[c * Cc + d] * inv - ((c == d) ? iden : 0.f);
        acc += vd * vd;
      }
    ortho = sqrtf(acc);
  }
  smc[threadIdx.x] = mc;
  sor[threadIdx.x] = ortho;
  __syncthreads();
  if (threadIdx.x == 0) {
    float a = 0.f, o = 0.f;
    for (int i = 0; i < Bc; ++i) { a += smc[i]; o += sor[i]; }
    accum[0] += a / (float)Bc;
    accum[1] += o / (float)Bc;
  }
}

// ---------------------------------------------------------------------------
// tail kernels
// ---------------------------------------------------------------------------
__global__ void cn_concat(const float* __restrict__ a, const float* __restrict__ b,
                          float* __restrict__ out, int rows, int Ca, int Cb) {
  int Ct = Ca + Cb;
  int t = blockIdx.x * blockDim.x + threadIdx.x;
  if (t >= rows * Ct) return;
  int c = t % Ct, r = t / Ct;
  out[t] = (c < Ca) ? a[(size_t)r * Ca + c] : b[(size_t)r * Cb + (c - Ca)];
}

__global__ void cn_max_axis1(const float* __restrict__ x, float* __restrict__ out,
                             int Bc, int n, int C) {
  int t = blockIdx.x * blockDim.x + threadIdx.x;
  if (t >= Bc * C) return;
  int c = t % C, b = t / C;
  const float* p = x + ((size_t)b * n) * C + c;
  float m = p[0];
  for (int i = 1; i < n; ++i) m = fmaxf(m, p[(size_t)i * C]);
  out[t] = m;
}

__global__ void cn_log_softmax(const float* __restrict__ x, float* __restrict__ out,
                               int rows, int C) {
  int r = blockIdx.x * blockDim.x + threadIdx.x;
  if (r >= rows) return;
  const float* p = x + (size_t)r * C;
  float m = p[0];
  for (int c = 1; c < C; ++c) m = fmaxf(m, p[c]);
  float s = 0.f;
  for (int c = 0; c < C; ++c) s += expf(p[c] - m);
  float ls = logf(s);
  float* o = out + (size_t)r * C;
  for (int c = 0; c < C; ++c) o[c] = p[c] - m - ls;
}

__global__ void cn_write_losses(const float* __restrict__ accum, float* __restrict__ out) {
  out[0] = accum[0];
  out[1] = accum[1];
}

// ---------------------------------------------------------------------------
// host orchestration
// ---------------------------------------------------------------------------
extern "C" void kernel_launch(void* const* d_in, const int* in_sizes, int n_in,
                              void* d_out, int out_size, void* d_ws, size_t ws_size,
                              hipStream_t stream) {
  (void)in_sizes; (void)n_in; (void)out_size; (void)ws_size;
  const int B = 8, N = 4096, KNN = 16;

  // Input order: depth-first over setup_inputs() dict insertion order:
  // x, conv1[0..2]{W,b,g,beta}, pool1{W,b}, conv2[0]{..}, pool2{W,b},
  // conv3[0..1]{..}, conv4[0..1]{..}, lin1[0]{..}, final[0..2]{..}
  const float* x0 = (const float*)d_in[0];
  const float* c1W[3] = {(const float*)d_in[1], (const float*)d_in[5], (const float*)d_in[9]};
  const float* c1b[3] = {(const float*)d_in[2], (const float*)d_in[6], (const float*)d_in[10]};
  const float* c1g[3] = {(const float*)d_in[3], (const float*)d_in[7], (const float*)d_in[11]};
  const float* c1e[3] = {(const float*)d_in[4], (const float*)d_in[8], (const float*)d_in[12]};
  const float* p1W = (const float*)d_in[13];
  const float* p1b = (const float*)d_in[14];
  const float* c2W = (const float*)d_in[15];
  const float* c2b = (const float*)d_in[16];
  const float* c2g = (const float*)d_in[17];
  const float* c2e = (const float*)d_in[18];
  const float* p2W = (const float*)d_in[19];
  const float* p2b = (const float*)d_in[20];
  const float* c3W[2] = {(const float*)d_in[21], (const float*)d_in[25]};
  const float* c3b[2] = {(const float*)d_in[22], (const float*)d_in[26]};
  const float* c3g[2] = {(const float*)d_in[23], (const float*)d_in[27]};
  const float* c3e[2] = {(const float*)d_in[24], (const float*)d_in[28]};
  const float* c4W[2] = {(const float*)d_in[29], (const float*)d_in[33]};
  const float* c4b[2] = {(const float*)d_in[30], (const float*)d_in[34]};
  const float* c4g[2] = {(const float*)d_in[31], (const float*)d_in[35]};
  const float* c4e[2] = {(const float*)d_in[32], (const float*)d_in[36]};
  const float* l1W = (const float*)d_in[37];
  const float* l1b = (const float*)d_in[38];
  const float* l1g = (const float*)d_in[39];
  const float* l1e = (const float*)d_in[40];
  const float* fW[3] = {(const float*)d_in[41], (const float*)d_in[45], (const float*)d_in[49]};
  const float* fb[3] = {(const float*)d_in[42], (const float*)d_in[46], (const float*)d_in[50]};
  const float* fg[3] = {(const float*)d_in[43], (const float*)d_in[47], (const float*)d_in[51]};
  const float* fe[3] = {(const float*)d_in[44], (const float*)d_in[48], (const float*)d_in[52]};

  // ---- workspace layout ----
  float* WS = (float*)d_ws;
  size_t off = 0;
  auto ALLOC = [&](size_t nf) { float* p = WS + off; off += nf; return p; };
  float* sq1 = ALLOC((size_t)B * N);
  float* dist = ALLOC((size_t)N * N);          // reused per batch
  int* nbr1 = (int*)ALLOC((size_t)B * N * KNN);
  float* ef1 = ALLOC((size_t)B * N * KNN * 6);
  float* bufA = ALLOC((size_t)B * N * KNN * 64);
  float* bufB = ALLOC((size_t)B * N * KNN * 64);
  float* x1 = ALLOC((size_t)B * N * 64);
  float* s1 = ALLOC((size_t)B * N * 32);
  float* deg1 = ALLOC((size_t)B * N);
  float* pool1 = ALLOC((size_t)B * 32 * 64);
  float* stats = ALLOC(2048);
  float* scshA = ALLOC(2048);
  float* scshB = ALLOC(2048);
  float* numb = ALLOC(B);
  float* denb = ALLOC(B);
  float* ss1 = ALLOC((size_t)B * 32 * 32);
  float* accum = ALLOC(2);
  int* nbr2 = (int*)ALLOC((size_t)B * 32 * 16);
  float* ef2 = ALLOC((size_t)B * 32 * 16 * 128);
  float* y2a = ALLOC((size_t)B * 32 * 16 * 128);
  float* x2 = ALLOC((size_t)B * 32 * 128);
  float* s2 = ALLOC((size_t)B * 32 * 8);
  float* deg2 = ALLOC((size_t)B * 32);
  float* pool2 = ALLOC((size_t)B * 8 * 128);
  float* ss2 = ALLOC((size_t)B * 8 * 8);
  int* nbr3 = (int*)ALLOC((size_t)B * 8 * 7);
  float* ef3 = ALLOC((size_t)448 * 512);
  float* y3a = ALLOC((size_t)448 * 512);
  float* y3b = ALLOC((size_t)448 * 512);
  float* x31 = ALLOC((size_t)B * 8 * 256);
  float* x32 = ALLOC((size_t)B * 8 * 512);
  float* cat3 = ALLOC((size_t)64 * 768);
  float* y4 = ALLOC((size_t)64 * 1024);
  float* gmax = ALLOC((size_t)B * 1024);
  float* f1 = ALLOC((size_t)B * 512);
  float* f2 = ALLOC((size_t)B * 256);
  float* f3 = ALLOC((size_t)B * 64);

  auto EWG = [](size_t n) { return dim3((unsigned)((n + 255) / 256)); };
  auto ZERO = [&](float* p, size_t nf) { cn_zero<<<EWG(nf), 256, 0, stream>>>(p, nf); };
  auto SHB = [](int Fin) {
    int Fin4 = (Fin + 3) & ~3;
    int KP = (Fin4 < 512) ? Fin4 : 512;
    return (size_t)(2 * Fin4 + KP * 16) * sizeof(float);
  };
  // GEMM + ReLU + BN-stats; inS = fused input-BN affine from previous layer (or null)
  auto GEMM_RS = [&](const float* Xp, const float* inS, const float* Wp, const float* bp,
                     float* Yp, int R, int Fin, int Fout, int MT) {
    ZERO(stats, 2 * (size_t)Fout);
    dim3 g((Fout + 15) / 16, (R + 16 * MT - 1) / (16 * MT));
    cn_gemm<true, true><<<g, 32, SHB(Fin), stream>>>(Xp, inS, Wp, bp, Yp, stats,
                                                     R, Fin, Fout, MT);
  };
  auto GEMM_PLAIN = [&](const float* Xp, const float* Wp, const float* bp, float* Yp,
                        int R, int Fin, int Fout, int MT) {
    dim3 g((Fout + 15) / 16, (R + 16 * MT - 1) / (16 * MT));
    cn_gemm<false, false><<<g, 32, SHB(Fin), stream>>>(Xp, nullptr, Wp, bp, Yp, nullptr,
                                                       R, Fin, Fout, MT);
  };
  auto FIN = [&](const float* gp, const float* ep, float* scshOut, size_t R, int Fout) {
    cn_bn_finalize<<<dim3((Fout + 63) / 64), 64, 0, stream>>>(stats, gp, ep, scshOut, Fout,
                                                              1.f / (float)R);
  };
  auto APPLY = [&](float* Yp, const float* scshIn, size_t R, int Fout) {
    size_t total = R * (size_t)Fout;
    cn_bn_apply<<<EWG(total), 256, 0, stream>>>(Yp, scshIn, total, Fout);
  };

  // ================= Stage 1 (N=4096, C=3) =================
  ZERO(accum, 2);
  cn_sqnorm<<<EWG((size_t)B * N), 256, 0, stream>>>(x0, sq1, B * N, 3);
  for (int b = 0; b < B; ++b) {
    cn_dist_wmma<<<dim3(N / 16, N / 16), 32, 0, stream>>>(x0, sq1, dist, b, N);
    cn_topk16<<<dim3(N), 32, 0, stream>>>(dist, nbr1, b, N, KNN);
  }
  const int E1 = B * N * KNN;
  cn_edge_feat<<<EWG((size_t)E1 * 6), 256, 0, stream>>>(x0, nbr1, ef1, B, N, KNN, 3);
  // conv1: BN of L1/L2 fused into the next layer's A-loads; only L3 applied.
  GEMM_RS(ef1, nullptr, c1W[0], c1b[0], bufA, E1, 6, 64, 32);   FIN(c1g[0], c1e[0], scshA, E1, 64);
  GEMM_RS(bufA, scshA, c1W[1], c1b[1], bufB, E1, 64, 64, 32);   FIN(c1g[1], c1e[1], scshB, E1, 64);
  GEMM_RS(bufB, scshB, c1W[2], c1b[2], bufA, E1, 64, 64, 32);   FIN(c1g[2], c1e[2], scshA, E1, 64);
  APPLY(bufA, scshA, E1, 64);
  cn_edge_max<<<EWG((size_t)B * N * 64), 256, 0, stream>>>(bufA, x1, B, N, KNN, 64);
  GEMM_PLAIN(x1, p1W, p1b, s1, B * N, 64, 32, 8);
  cn_softmax_rows<<<EWG((size_t)B * N), 256, 0, stream>>>(s1, B * N, 32);
  cn_pool_stx<<<dim3(64 / 16, 32 / 16, B), 32, 0, stream>>>(s1, x1, pool1, N, 32, 64);
  ZERO(deg1, (size_t)B * N);
  cn_deg<<<EWG((size_t)E1), 256, 0, stream>>>(nbr1, deg1, B, N, KNN);
  ZERO(numb, B); ZERO(denb, B);
  cn_mincut_num<<<EWG((size_t)E1), 256, 0, stream>>>(s1, nbr1, numb, B, N, KNN, 32);
  cn_mincut_den<<<EWG((size_t)B * N), 256, 0, stream>>>(s1, deg1, denb, B, N, 32);
  cn_ss<<<EWG((size_t)B * 32 * 32), 256, 0, stream>>>(s1, ss1, B, N, 32);
  cn_mincut_loss<<<1, 32, 0, stream>>>(numb, denb, ss1, accum, B, 32);

  // ================= Stage 2 (n=32, C=64) =================
  cn_knn_small<<<dim3(B), 256, 0, stream>>>(pool1, nbr2, 32, 64, 16);
  const int E2 = B * 32 * 16;
  cn_edge_feat<<<EWG((size_t)E2 * 128), 256, 0, stream>>>(pool1, nbr2, ef2, B, 32, 16, 64);
  GEMM_RS(ef2, nullptr, c2W, c2b, y2a, E2, 128, 128, 8);  FIN(c2g, c2e, scshA, E2, 128);
  APPLY(y2a, scshA, E2, 128);
  cn_edge_max<<<EWG((size_t)B * 32 * 128), 256, 0, stream>>>(y2a, x2, B, 32, 16, 128);
  GEMM_PLAIN(x2, p2W, p2b, s2, B * 32, 128, 8, 1);
  cn_softmax_rows<<<EWG((size_t)B * 32), 256, 0, stream>>>(s2, B * 32, 8);
  cn_pool_stx<<<dim3(128 / 16, 1, B), 32, 0, stream>>>(s2, x2, pool2, 32, 8, 128);
  ZERO(deg2, (size_t)B * 32);
  cn_deg<<<EWG((size_t)E2), 256, 0, stream>>>(nbr2, deg2, B, 32, 16);
  ZERO(numb, B); ZERO(denb, B);
  cn_mincut_num<<<EWG((size_t)E2), 256, 0, stream>>>(s2, nbr2, numb, B, 32, 16, 8);
  cn_mincut_den<<<EWG((size_t)B * 32), 256, 0, stream>>>(s2, deg2, denb, B, 32, 8);
  cn_ss<<<EWG((size_t)B * 8 * 8), 256, 0, stream>>>(s2, ss2, B, 32, 8);
  cn_mincut_loss<<<1, 32, 0, stream>>>(numb, denb, ss2, accum, B, 8);

  // ================= Stage 3 (n=8, k=7=n-1) =================
  const int E3 = B * 8 * 7;  // 448
  cn_nbr_all<<<EWG((size_t)E3), 256, 0, stream>>>(nbr3, B, 8, 7);
  cn_edge_feat<<<EWG((size_t)E3 * 256), 256, 0, stream>>>(pool2, nbr3, ef3, B, 8, 7, 128);
  GEMM_RS(ef3, nullptr, c3W[0], c3b[0], y3a, E3, 256, 256, 1); FIN(c3g[0], c3e[0], scshA, E3, 256);
  GEMM_RS(y3a, scshA, c3W[1], c3b[1], y3b, E3, 256, 256, 1);   FIN(c3g[1], c3e[1], scshB, E3, 256);
  APPLY(y3b, scshB, E3, 256);
  cn_edge_max<<<EWG((size_t)B * 8 * 256), 256, 0, stream>>>(y3b, x31, B, 8, 7, 256);
  cn_edge_feat<<<EWG((size_t)E3 * 512), 256, 0, stream>>>(x31, nbr3, ef3, B, 8, 7, 256);
  GEMM_RS(ef3, nullptr, c4W[0], c4b[0], y3a, E3, 512, 512, 1); FIN(c4g[0], c4e[0], scshA, E3, 512);
  GEMM_RS(y3a, scshA, c4W[1], c4b[1], y3b, E3, 512, 512, 1);   FIN(c4g[1], c4e[1], scshB, E3, 512);
  APPLY(y3b, scshB, E3, 512);
  cn_edge_max<<<EWG((size_t)B * 8 * 512), 256, 0, stream>>>(y3b, x32, B, 8, 7, 512);
  cn_concat<<<EWG((size_t)64 * 768), 256, 0, stream>>>(x31, x32, cat3, 64, 256, 512);
  GEMM_RS(cat3, nullptr, l1W, l1b, y4, 64, 768, 1024, 1);      FIN(l1g, l1e, scshA, 64, 1024);
  APPLY(y4, scshA, 64, 1024);
  cn_max_axis1<<<EWG((size_t)B * 1024), 256, 0, stream>>>(y4, gmax, B, 8, 1024);
  GEMM_RS(gmax, nullptr, fW[0], fb[0], f1, 8, 1024, 512, 1);   FIN(fg[0], fe[0], scshA, 8, 512);
  GEMM_RS(f1, scshA, fW[1], fb[1], f2, 8, 512, 256, 1);        FIN(fg[1], fe[1], scshB, 8, 256);
  GEMM_RS(f2, scshB, fW[2], fb[2], f3, 8, 256, 40, 1);         FIN(fg[2], fe[2], scshA, 8, 40);
  APPLY(f3, scshA, 8, 40);
  cn_log_softmax<<<dim3(1), 32, 0, stream>>>(f3, (float*)d_out, 8, 40);
  cn_write_losses<<<1, 1, 0, stream>>>(accum, (float*)d_out + 320);
}